// LieFucWithBracketWeightOptimized_30855045054563
// MI455X (gfx1250) — compile-verified
//
#include <hip/hip_runtime.h>
#include <hip/hip_bf16.h>

// Problem constants (from reference): B=8, D=64, S=512.
#define PB 8
#define PD 64
#define PS 512

typedef __attribute__((ext_vector_type(2))) float v2f;
typedef __attribute__((ext_vector_type(8))) float v8f;

__device__ __forceinline__ v8f wmma_f32_16x16x4(v2f a, v2f b, v8f c) {
    // D = A(16x4, f32) * B(4x16, f32) + C(16x16, f32)
    return __builtin_amdgcn_wmma_f32_16x16x4_f32(
        /*neg_a=*/false, a, /*neg_b=*/false, b,
        /*c_mod=*/(short)0, c, /*reuse_a=*/false, /*reuse_b=*/false);
}

__device__ __forceinline__ v8f v8f_zero() {
    v8f z = {0.f, 0.f, 0.f, 0.f, 0.f, 0.f, 0.f, 0.f};
    return z;
}

// ---------------------------------------------------------------------------
// Stage 1: A = x^T * W_w  and  Bm = x^T * W_h   (per batch: 512x64 each)
//   A[b][j][k]  = sum_d x[b][d][j] * W[d][k]        (W_w = W rows 0..63)
//   Bm[b][j][k] = sum_d x[b][d][j] * W[64+d][k]     (W_h = W rows 64..127)
// One wave handles one (b, 16-row j-strip): 4 N-tiles x 2 outputs =
// 8 fp32 WMMA accumulators, K=64 in 16 steps of 4.
// Grid: 64 blocks x 128 threads (4 waves) = 256 waves = 8 batches * 32 strips.
// ---------------------------------------------------------------------------
__global__ __launch_bounds__(128) void lie_stage1(
    const float* __restrict__ x, const float* __restrict__ W,
    float* __restrict__ Aws, float* __restrict__ Bws) {

    const int lane = threadIdx.x & 31;
    const int wv   = threadIdx.x >> 5;
    const int gw   = blockIdx.x * 4 + wv;    // 0..255
    const int b    = gw >> 5;                // 0..7
    const int jb   = (gw & 31) * 16;         // j strip base
    const int ln   = lane & 15;
    const int kh   = lane >> 4;              // 0 or 1 (half-lane group)

    // L2 prefetch of the (tiny, fully reused) weight matrix.
    __builtin_prefetch(W + threadIdx.x * 4, 0, 1);

    const float* xb = x + (size_t)b * PD * PS;

    v8f accA[4], accB[4];
#pragma unroll
    for (int nt = 0; nt < 4; ++nt) { accA[nt] = v8f_zero(); accB[nt] = v8f_zero(); }

#pragma unroll
    for (int kb = 0; kb < PD; kb += 4) {
        const int k0 = kb + 2 * kh;          // ISA layout: lanes 0-15 hold K={kb,kb+1}, lanes 16-31 K={kb+2,kb+3}
        // A-operand tile: A_op[m][k] = x[b][k][jb+m]  (m = ln)
        v2f a;
        a.x = xb[(size_t)k0 * PS + jb + ln];
        a.y = xb[(size_t)(k0 + 1) * PS + jb + ln];
#pragma unroll
        for (int nt = 0; nt < 4; ++nt) {
            const int n = nt * 16 + ln;
            v2f bw, bh;
            bw.x = W[(size_t)k0 * PD + n];
            bw.y = W[(size_t)(k0 + 1) * PD + n];
            bh.x = W[(size_t)(k0 + PD) * PD + n];
            bh.y = W[(size_t)(k0 + 1 + PD) * PD + n];
            accA[nt] = wmma_f32_16x16x4(a, bw, accA[nt]);
            accB[nt] = wmma_f32_16x16x4(a, bh, accB[nt]);
        }
    }

    // Store D tiles: lane holds rows (v + 8*kh), col = nt*16 + ln.
    float* Ab = Aws + ((size_t)b * PS + jb) * PD;
    float* Bb = Bws + ((size_t)b * PS + jb) * PD;
#pragma unroll
    for (int nt = 0; nt < 4; ++nt) {
#pragma unroll
        for (int v = 0; v < 8; ++v) {
            const int r = v + 8 * kh;
            Ab[(size_t)r * PD + nt * 16 + ln] = accA[nt][v];
            Bb[(size_t)r * PD + nt * 16 + ln] = accB[nt][v];
        }
    }
}

// ---------------------------------------------------------------------------
// Stage 2: per (batch, 64-wide i-tile):
//   wmat[w][i] = (i <= w) ? 1 - sqrt(|A_i|^2 + |B_w|^2 + 2*G[w][i]) : 1
//       with G = Bm * A^T   (WMMA GEMM #1, per 64-w chunk)
//   c[d][i]   += x[d][w] * wmat[w][i]             (WMMA GEMM #2, accumulated)
// 8 waves; each wave owns a 16x32 output patch: mt = (wv&3)*16 rows,
// nh = (wv>>2)*32 cols (two 16-col WMMA tiles). Same patching reused for G.
// ---------------------------------------------------------------------------
__global__ __launch_bounds__(256) void lie_stage2(
    const float* __restrict__ x, const float* __restrict__ Aws,
    const float* __restrict__ Bws, float* __restrict__ out) {

    const int b  = blockIdx.y;           // batch
    const int ib = blockIdx.x * 64;      // i-tile base

    __shared__ float sA[64][65];         // A rows of this i-tile   (odd stride: no LDS bank conflicts)
    __shared__ float sB[64][65];         // Bm rows of current w-chunk
    __shared__ float sWm[64][64];        // wmat chunk [w_local][i_local]
    __shared__ float sNa[64];            // |A_i|^2
    __shared__ float sNb[64];            // |B_w|^2 (per chunk)

    const int t    = threadIdx.x;
    const int lane = t & 31;
    const int wv   = t >> 5;
    const int ln   = lane & 15;
    const int kh   = lane >> 4;
    const int mt   = (wv & 3) * 16;      // row-tile base (w for G, d for c)
    const int nh   = (wv >> 2) * 32;     // col-half base (i)

    // Load A rows for this i-tile (coalesced), compute row norms.
    const float* Ab = Aws + ((size_t)b * PS + ib) * PD;
#pragma unroll
    for (int p = 0; p < 16; ++p) {
        const int lin = t + 256 * p;     // 0..4095
        sA[lin >> 6][lin & 63] = Ab[lin];
    }
    __syncthreads();
    if (t < 64) {
        float s = 0.f;
#pragma unroll
        for (int k = 0; k < 64; ++k) { const float v = sA[t][k]; s += v * v; }
        sNa[t] = s;
    }
    __syncthreads();

    const float* xb = x + (size_t)b * PD * PS;
    v8f c0 = v8f_zero(), c1 = v8f_zero();    // persistent 16x32 c accumulator

    for (int wc = 0; wc < PS; wc += 64) {    // 8 chunks over w
        // Stage Bm chunk into LDS (coalesced).
        const float* Bb = Bws + ((size_t)b * PS + wc) * PD;
#pragma unroll
        for (int p = 0; p < 16; ++p) {
            const int lin = t + 256 * p;
            sB[lin >> 6][lin & 63] = Bb[lin];
        }
        __syncthreads();
        if (t < 64) {
            float s = 0.f;
#pragma unroll
            for (int k = 0; k < 64; ++k) { const float v = sB[t][k]; s += v * v; }
            sNb[t] = s;
        }
        __syncthreads();

        // GEMM #1: G[w][i] = sum_k sB[w][k] * sA[i][k]  (A-op = sB, B-op = sA^T)
        v8f g0 = v8f_zero(), g1 = v8f_zero();
#pragma unroll
        for (int kb = 0; kb < 64; kb += 4) {
            const int k0 = kb + 2 * kh;
            v2f a;  a.x  = sB[mt + ln][k0];        a.y  = sB[mt + ln][k0 + 1];
            v2f b0; b0.x = sA[nh + ln][k0];        b0.y = sA[nh + ln][k0 + 1];
            v2f b1; b1.x = sA[nh + 16 + ln][k0];   b1.y = sA[nh + 16 + ln][k0 + 1];
            g0 = wmma_f32_16x16x4(a, b0, g0);
            g1 = wmma_f32_16x16x4(a, b1, g1);
        }

        // Transform each lane's own D registers into wmat (sqrt + tril mask).
#pragma unroll
        for (int v = 0; v < 8; ++v) {
            const int wrow = mt + v + 8 * kh;      // local w of this D element
            const int gwi  = wc + wrow;            // global w
            const int i0   = nh + ln;
            const int i1   = nh + 16 + ln;
            float m0 = 1.0f, m1 = 1.0f;
            if (ib + i0 <= gwi)
                m0 = 1.0f - sqrtf(fmaxf(sNa[i0] + sNb[wrow] + 2.0f * g0[v], 0.0f));
            if (ib + i1 <= gwi)
                m1 = 1.0f - sqrtf(fmaxf(sNa[i1] + sNb[wrow] + 2.0f * g1[v], 0.0f));
            sWm[wrow][i0] = m0;
            sWm[wrow][i1] = m1;
        }
        __syncthreads();

        // GEMM #2: c[d][i] += sum_w x[d][wc+w] * sWm[w][i]
#pragma unroll
        for (int kb = 0; kb < 64; kb += 4) {
            const int k0 = kb + 2 * kh;
            v2f a;  a.x  = xb[(size_t)(mt + ln) * PS + wc + k0];
                    a.y  = xb[(size_t)(mt + ln) * PS + wc + k0 + 1];
            v2f b0; b0.x = sWm[k0][nh + ln];       b0.y = sWm[k0 + 1][nh + ln];
            v2f b1; b1.x = sWm[k0][nh + 16 + ln];  b1.y = sWm[k0 + 1][nh + 16 + ln];
            c0 = wmma_f32_16x16x4(a, b0, c0);
            c1 = wmma_f32_16x16x4(a, b1, c1);
        }
        __syncthreads();   // protect sB/sWm before next chunk
    }

    // Store c tile: lane holds rows d = mt + v + 8*kh, cols ib+nh+ln (+16).
    float* ob = out + (size_t)b * PD * PS;
#pragma unroll
    for (int v = 0; v < 8; ++v) {
        const int d = mt + v + 8 * kh;
        ob[(size_t)d * PS + ib + nh + ln]      = c0[v];
        ob[(size_t)d * PS + ib + nh + 16 + ln] = c1[v];
    }
}

extern "C" void kernel_launch(void* const* d_in, const int* in_sizes, int n_in,
                              void* d_out, int out_size, void* d_ws, size_t ws_size,
                              hipStream_t stream) {
    const float* x = (const float*)d_in[0];   // (8, 64, 512) f32
    const float* W = (const float*)d_in[1];   // (128, 64)    f32
    float* out = (float*)d_out;               // (8, 64, 512) f32

    float* Aws = (float*)d_ws;                        // 8*512*64 f32 = 1 MB
    float* Bws = Aws + (size_t)PB * PS * PD;          // 1 MB more

    lie_stage1<<<64, 128, 0, stream>>>(x, W, Aws, Bws);
    lie_stage2<<<dim3(PS / 64, PB), 256, 0, stream>>>(x, Aws, Bws, out);
}